// GraphAttentionLayer_50749333569597
// MI455X (gfx1250) — compile-verified
//
#include <hip/hip_runtime.h>
#include <hip/hip_bf16.h>
#include <math.h>

#define HID 128

typedef __attribute__((ext_vector_type(2))) float v2f;
typedef __attribute__((ext_vector_type(8))) float v8f;

// ---- ordered-uint encoding for float atomic max (handles negatives) ----
__device__ __forceinline__ unsigned f2ord(float f) {
  unsigned i = __float_as_uint(f);
  return (i & 0x80000000u) ? ~i : (i | 0x80000000u);
}
__device__ __forceinline__ float ord2f(unsigned u) {
  return __uint_as_float((u & 0x80000000u) ? (u & 0x7fffffffu) : ~u);
}

// ============================================================
// Kernel 1: h = X @ W^T via V_WMMA_F32_16X16X4_F32.
// One wave computes a 16x128 tile of h. W^T staged in 64KB LDS.
// A frag (16x4 f32): lane m = lane&15; vgpr0 = X[m][kk+2*hi], vgpr1 = X[m][kk+2*hi+1]
// B frag (4x16 f32): lane n = lane&15; vgpr0 = Wt[kk+2*hi][n], vgpr1 = Wt[kk+2*hi+1][n]
// C/D: vgpr v -> row (v + 8*hi), col = lane&15.
// Epilogue: single full-tile check -> branch-free coalesced stores.
// ============================================================
__global__ void __launch_bounds__(256)
gemm_proj_kernel(const float* __restrict__ X, const float* __restrict__ W,
                 float* __restrict__ H, int N) {
  __shared__ float Wt[HID * HID];  // Wt[k*HID + n] = W[n*HID + k]
  for (int i = threadIdx.x; i < HID * HID; i += blockDim.x) {
    int n = i >> 7;
    int k = i & (HID - 1);
    Wt[k * HID + n] = W[i];
  }
  __syncthreads();

  int wave  = threadIdx.x >> 5;
  int lane  = threadIdx.x & 31;
  int mtile = blockIdx.x * (blockDim.x >> 5) + wave;
  int row0  = mtile * 16;
  if (row0 >= N) return;

  int mr = lane & 15;
  int hi = lane >> 4;
  int arow = row0 + mr;
  if (arow >= N) arow = N - 1;  // tail clamp for A reads (duplicate rows OK)
  const float* xrow = X + (size_t)arow * HID;

  v8f acc[8] = {};
  for (int kk = 0; kk < HID; kk += 4) {
    int ka = kk + 2 * hi;
    v2f a;
    a.x = xrow[ka];
    a.y = xrow[ka + 1];
#pragma unroll
    for (int nt = 0; nt < 8; ++nt) {
      int n = nt * 16 + mr;
      v2f b;
      b.x = Wt[ka * HID + n];
      b.y = Wt[(ka + 1) * HID + n];
      acc[nt] = __builtin_amdgcn_wmma_f32_16x16x4_f32(
          false, a, false, b, (short)0, acc[nt], false, false);
    }
  }

  // Store: fast path when the whole 16-row tile is in range (always true
  // when N % 16 == 0, as here) -> no per-element exec-mask churn.
  float* hbase = H + (size_t)(row0 + 8 * hi) * HID + mr;
  if (row0 + 16 <= N) {
#pragma unroll
    for (int nt = 0; nt < 8; ++nt) {
#pragma unroll
      for (int v = 0; v < 8; ++v) {
        hbase[(size_t)v * HID + nt * 16] = acc[nt][v];
      }
    }
  } else {
#pragma unroll
    for (int nt = 0; nt < 8; ++nt) {
#pragma unroll
      for (int v = 0; v < 8; ++v) {
        int r = row0 + v + 8 * hi;
        if (r < N) H[(size_t)r * HID + nt * 16 + mr] = acc[nt][v];
      }
    }
  }
}

// ============================================================
// Kernel 2: a_src[i] = h[i]·w_edge[0:128], a_dst[i] = h[i]·w_edge[128:256]
// One wave per node row; float4 per lane; wave32 shuffle reduce.
// ============================================================
__global__ void avec_kernel(const float* __restrict__ H, const float* __restrict__ we,
                            float* __restrict__ a_src, float* __restrict__ a_dst, int N) {
  int w = (blockIdx.x * blockDim.x + threadIdx.x) >> 5;
  int lane = threadIdx.x & 31;
  if (w >= N) return;
  float4 h4 = ((const float4*)(H + (size_t)w * HID))[lane];
  float4 ws = ((const float4*)we)[lane];
  float4 wd = ((const float4*)(we + HID))[lane];
  float ps = h4.x * ws.x + h4.y * ws.y + h4.z * ws.z + h4.w * ws.w;
  float pd = h4.x * wd.x + h4.y * wd.y + h4.z * wd.z + h4.w * wd.w;
#pragma unroll
  for (int m = 16; m > 0; m >>= 1) {
    ps += __shfl_xor(ps, m, 32);
    pd += __shfl_xor(pd, m, 32);
  }
  if (lane == 0) { a_src[w] = ps; a_dst[w] = pd; }
}

// ============================================================
// Kernel 3: zero scratch region (agg + denom + maxkey)
// ============================================================
__global__ void zero_kernel(float* __restrict__ p, size_t n) {
  size_t i = blockIdx.x * (size_t)blockDim.x + threadIdx.x;
  size_t stride = gridDim.x * (size_t)blockDim.x;
  for (; i < n; i += stride) p[i] = 0.0f;
}

// ============================================================
// Kernel 4: per-edge leaky_relu score + global max (ordered-uint atomicMax,
// one atomic per wave after shuffle-max)
// ============================================================
__global__ void edge_score_kernel(const int* __restrict__ src, const int* __restrict__ dst,
                                  const float* __restrict__ a_src, const float* __restrict__ a_dst,
                                  float* __restrict__ sbuf, unsigned* __restrict__ maxkey, int E) {
  int i = blockIdx.x * blockDim.x + threadIdx.x;
  unsigned key = 0u;  // minimum ordered key
  if (i < E) {
    float s = a_src[src[i]] + a_dst[dst[i]];
    s = (s > 0.0f) ? s : 0.2f * s;  // leaky_relu, slope 0.2
    sbuf[i] = s;
    key = f2ord(s);
  }
#pragma unroll
  for (int m = 16; m > 0; m >>= 1) {
    unsigned o = __shfl_xor(key, m, 32);
    key = (o > key) ? o : key;
  }
  if ((threadIdx.x & 31) == 0) atomicMax(maxkey, key);
}

// ============================================================
// Kernel 5: attn = exp(s - max); denom[dst] += attn (native f32 atomics)
// ============================================================
__global__ void edge_exp_kernel(const int* __restrict__ dst, float* __restrict__ sbuf,
                                const unsigned* __restrict__ maxkey,
                                float* __restrict__ denom, int E) {
  int i = blockIdx.x * blockDim.x + threadIdx.x;
  if (i >= E) return;
  float mx = ord2f(*maxkey);
  float a = expf(sbuf[i] - mx);
  sbuf[i] = a;
  unsafeAtomicAdd(&denom[dst[i]], a);
}

// ============================================================
// Kernel 6: scatter  agg[dst] += h[src] * (attn / (denom[dst]+1e-6))
// One wave per edge, float4 per lane, native global_atomic_add_f32.
// h and agg are L2-resident (25.6MB each vs 192MB L2), so the ~800MB of
// gather+atomic traffic stays on-die; HBM sees only ~150MB total.
// ============================================================
__global__ void scatter_kernel(const int* __restrict__ src, const int* __restrict__ dst,
                               const float* __restrict__ sbuf, const float* __restrict__ denom,
                               const float* __restrict__ H, float* __restrict__ agg, int E) {
  long long gid = blockIdx.x * (long long)blockDim.x + threadIdx.x;
  long long nwaves = (gridDim.x * (long long)blockDim.x) >> 5;
  int lane = threadIdx.x & 31;
  for (long long e = gid >> 5; e < E; e += nwaves) {
    int s = src[e], d = dst[e];
    float w = sbuf[e] / (denom[d] + 1e-6f);
    float4 h4 = ((const float4*)(H + (size_t)s * HID))[lane];
    float* ap = agg + (size_t)d * HID + lane * 4;
    unsafeAtomicAdd(ap + 0, h4.x * w);
    unsafeAtomicAdd(ap + 1, h4.y * w);
    unsafeAtomicAdd(ap + 2, h4.z * w);
    unsafeAtomicAdd(ap + 3, h4.w * w);
  }
}

// ============================================================
// Kernel 7: out = gelu_exact(layernorm(agg + x) * gamma + beta)
// One wave per row; wave32 shuffle reductions for mean/var.
// ============================================================
__global__ void final_kernel(const float* __restrict__ agg, const float* __restrict__ X,
                             const float* __restrict__ gamma, const float* __restrict__ beta,
                             float* __restrict__ out, int N) {
  int w = (blockIdx.x * blockDim.x + threadIdx.x) >> 5;
  int lane = threadIdx.x & 31;
  if (w >= N) return;
  float4 a4 = ((const float4*)(agg + (size_t)w * HID))[lane];
  float4 x4 = ((const float4*)(X + (size_t)w * HID))[lane];
  float v0 = a4.x + x4.x, v1 = a4.y + x4.y, v2 = a4.z + x4.z, v3 = a4.w + x4.w;

  float s = v0 + v1 + v2 + v3;
#pragma unroll
  for (int m = 16; m > 0; m >>= 1) s += __shfl_xor(s, m, 32);
  float mu = s * (1.0f / HID);

  float d0 = v0 - mu, d1 = v1 - mu, d2 = v2 - mu, d3 = v3 - mu;
  float q = d0 * d0 + d1 * d1 + d2 * d2 + d3 * d3;
#pragma unroll
  for (int m = 16; m > 0; m >>= 1) q += __shfl_xor(q, m, 32);
  float rs = rsqrtf(q * (1.0f / HID) + 1e-5f);

  float4 g4 = ((const float4*)gamma)[lane];
  float4 b4 = ((const float4*)beta)[lane];
  float y0 = d0 * rs * g4.x + b4.x;
  float y1 = d1 * rs * g4.y + b4.y;
  float y2 = d2 * rs * g4.z + b4.z;
  float y3 = d3 * rs * g4.w + b4.w;

  const float k = 0.7071067811865476f;
  float4 o;
  o.x = 0.5f * y0 * (1.0f + erff(y0 * k));
  o.y = 0.5f * y1 * (1.0f + erff(y1 * k));
  o.z = 0.5f * y2 * (1.0f + erff(y2 * k));
  o.w = 0.5f * y3 * (1.0f + erff(y3 * k));
  ((float4*)(out + (size_t)w * HID))[lane] = o;
}

extern "C" void kernel_launch(void* const* d_in, const int* in_sizes, int n_in,
                              void* d_out, int out_size, void* d_ws, size_t ws_size,
                              hipStream_t stream) {
  const float* X     = (const float*)d_in[0];   // node_states [N,128]
  const int*   EI    = (const int*)d_in[1];     // edge_index  [2,E]
  const float* W     = (const float*)d_in[2];   // W_proj [128,128]
  const float* wedge = (const float*)d_in[3];   // w_edge [256]
  const float* gamma = (const float*)d_in[4];
  const float* beta  = (const float*)d_in[5];

  int N = in_sizes[0] / HID;
  int E = in_sizes[1] / 2;
  const int* src = EI;
  const int* dst = EI + E;

  // workspace layout: h | agg | denom | maxkey | a_src | a_dst | sbuf
  float*    h      = (float*)d_ws;
  float*    agg    = h + (size_t)N * HID;
  float*    denom  = agg + (size_t)N * HID;
  unsigned* maxkey = (unsigned*)(denom + N);
  float*    a_src  = (float*)(maxkey + 1);
  float*    a_dst  = a_src + N;
  float*    sbuf   = a_dst + N;

  // zero agg + denom + maxkey (contiguous)
  size_t nz = (size_t)N * HID + (size_t)N + 1;
  zero_kernel<<<1024, 256, 0, stream>>>(agg, nz);

  int mtiles = (N + 15) / 16;
  gemm_proj_kernel<<<(mtiles + 7) / 8, 256, 0, stream>>>(X, W, h, N);

  avec_kernel<<<((size_t)N * 32 + 255) / 256, 256, 0, stream>>>(h, wedge, a_src, a_dst, N);

  edge_score_kernel<<<(E + 255) / 256, 256, 0, stream>>>(src, dst, a_src, a_dst, sbuf, maxkey, E);

  edge_exp_kernel<<<(E + 255) / 256, 256, 0, stream>>>(dst, sbuf, maxkey, denom, E);

  long long scatter_threads = (long long)E * 32;
  int scatter_blocks = (int)((scatter_threads + 255) / 256);
  if (scatter_blocks > 200000) scatter_blocks = 200000;
  scatter_kernel<<<scatter_blocks, 256, 0, stream>>>(src, dst, sbuf, denom, h, agg, E);

  final_kernel<<<((size_t)N * 32 + 255) / 256, 256, 0, stream>>>(agg, X, gamma, beta, (float*)d_out, N);
}